// Top2gate_46883863003247
// MI455X (gfx1250) — compile-verified
//
#include <hip/hip_runtime.h>
#include <hip/hip_bf16.h>

typedef __attribute__((ext_vector_type(2))) float v2f;
typedef __attribute__((ext_vector_type(8))) float v8f;

#define N_EMBD   2048
#define N_EXP    8
#define N_TOK    16384          // 4 * 4096
#define KSLICES  8              // waves per tile (block) splitting the K dim
#define KS       (N_EMBD / KSLICES)   // 256 K per wave
#define W_STRIDE 2052           // LDS row stride (floats): %64==4 -> conflict-free, 16B aligned

// wave32 xor-shuffle via ds_swizzle (group-of-32 mode: and=0x1f, or=0, xor=M)
template <int M>
__device__ __forceinline__ float swzXorF(float v) {
    return __int_as_float(
        __builtin_amdgcn_ds_swizzle(__float_as_int(v), (M << 10) | 0x1f));
}
template <int M>
__device__ __forceinline__ int swzXorI(int v) {
    return __builtin_amdgcn_ds_swizzle(v, (M << 10) | 0x1f);
}

__device__ __forceinline__ float groupSum8(float v) {
    v += swzXorF<1>(v);
    v += swzXorF<2>(v);
    v += swzXorF<4>(v);
    return v;
}

// branchless max+argmax over this lane's xor-group of 8; lowest index wins ties
template <int M>
__device__ __forceinline__ void amStep(float& v, int& idx) {
    float ov = swzXorF<M>(v);
    int   oi = swzXorI<M>(idx);
    bool take = (ov > v) | ((ov == v) & (oi < idx));
    v   = take ? ov : v;
    idx = take ? oi : idx;
}
__device__ __forceinline__ void groupArgMax8(float& v, int& idx) {
    amStep<1>(v, idx);
    amStep<2>(v, idx);
    amStep<4>(v, idx);
}

__global__ __launch_bounds__(256) void top2gate_kernel(
        const float* __restrict__ x, const float* __restrict__ W,
        float* __restrict__ out_g, int* __restrict__ out_i,
        float* __restrict__ ws_imp, float* __restrict__ ws_load) {
    __shared__ float Wlds[N_EXP * W_STRIDE];        // W row-major, padded rows
    __shared__ float Pds[(KSLICES - 1) * 32 * 8];   // partial accumulators

    const int tid = threadIdx.x;

    // vectorized W fill: 4096 float4s, 16 per thread, coalesced both sides
    const float4* W4 = (const float4*)W;
    for (int j = tid; j < N_EXP * (N_EMBD / 4); j += 256) {
        int e  = j >> 9;          // 512 float4 per expert row
        int k4 = j & 511;
        *(float4*)&Wlds[e * W_STRIDE + k4 * 4] = W4[j];
    }
    __syncthreads();

    const int lane  = tid & 31;
    const int wave  = tid >> 5;          // = K-slice index
    const int tile  = blockIdx.x;        // 16 tokens per block
    const int eidx  = lane & 15;         // A row in tile / B column (expert)
    const bool evalid = (eidx < N_EXP);
    const int off   = (lane < 16) ? 0 : 2;
    const int kbase = wave * KS;

    const float* xrow = x + (size_t)(tile * 16 + eidx) * N_EMBD + kbase;
    const int    wbase = (eidx & 7) * W_STRIDE + kbase;

    // D = A(16x4 f32) * B(4x16 f32) + C over this wave's K slice.
    // unroll 16 -> clause of 16 NT global_load_b64 in flight per wave,
    // covering the HBM BW*latency product at 8 waves/SIMD.
    v8f c = {};
#pragma unroll 16
    for (int k = 0; k < KS; k += 4) {
        v2f a = __builtin_nontemporal_load((const v2f*)(xrow + k + off));
        float b0 = Wlds[wbase + k + off];
        float b1 = Wlds[wbase + k + off + 1];
        v2f b;
        b.x = evalid ? b0 : 0.0f;    // pad experts 8..15 -> zero columns
        b.y = evalid ? b1 : 0.0f;
        c = __builtin_amdgcn_wmma_f32_16x16x4_f32(
                false, a, false, b, (short)0, c, false, false);
    }

    // combine K-slice partials: waves 1..7 dump to LDS, wave 0 reduces
    if (wave != 0) {
        int base = ((wave - 1) * 32 + lane) * 8;
#pragma unroll
        for (int i = 0; i < 8; ++i) Pds[base + i] = c[i];
    }
    __syncthreads();
    if (wave != 0) return;

#pragma unroll
    for (int s = 0; s < KSLICES - 1; ++s) {
        int base = (s * 32 + lane) * 8;
#pragma unroll
        for (int i = 0; i < 8; ++i) c[i] += Pds[base + i];
    }

    // c[i], lanes 0-15 : logits[token i    ][expert = lane]
    // c[i], lanes 16-31: logits[token 8 + i][expert = lane-16]
    float sum_imp  = 0.0f;
    float cnt_load = 0.0f;
    const int tokBase = tile * 16 + ((lane >> 4) << 3);

#pragma unroll
    for (int i = 0; i < 8; ++i) {
        float logit = c[i];

        float v1 = evalid ? logit : -3.4e38f;
        int   i1 = eidx;
        groupArgMax8(v1, i1);                 // max logit + top-1 index

        float p = evalid ? __expf(logit - v1) : 0.0f;
        float s = groupSum8(p);
        float prob = evalid ? (p / s) : 0.0f; // full softmax prob
        sum_imp  += prob;
        cnt_load += (evalid && eidx == i1) ? 1.0f : 0.0f;

        float v2 = (evalid && eidx != i1) ? logit : -3.4e38f;
        int   i2 = eidx;
        groupArgMax8(v2, i2);                 // second-best

        if (eidx == 0) {                      // lanes 0 and 16 write their token
            float e2 = __expf(v2 - v1);       // p2/p1
            float r  = 1.0f / (1.0f + e2);
            int tok = tokBase + i;
            *(float2*)(out_g + 2 * tok) = make_float2(r, e2 * r);
            *(int2*)  (out_i + 2 * tok) = make_int2(i1, i2);
        }
    }

    // fold the two half-wave token groups, then one atomic per expert per wave
    sum_imp  += swzXorF<16>(sum_imp);
    cnt_load += swzXorF<16>(cnt_load);
    if (lane < N_EXP) {
        atomicAdd(&ws_imp[lane],  sum_imp);
        atomicAdd(&ws_load[lane], cnt_load);
    }
}

__global__ void top2gate_finalize(const float* __restrict__ ws_imp,
                                  const float* __restrict__ ws_load,
                                  float* __restrict__ out_aux) {
    if (threadIdx.x == 0 && blockIdx.x == 0) {
        const float invN = 1.0f / (float)N_TOK;
        float aux = 0.0f;
#pragma unroll
        for (int e = 0; e < N_EXP; ++e)
            aux += (ws_imp[e] * invN) * (ws_load[e] * invN);
        out_aux[0] = (float)N_EXP * 0.01f * aux;
    }
}

extern "C" void kernel_launch(void* const* d_in, const int* in_sizes, int n_in,
                              void* d_out, int out_size, void* d_ws, size_t ws_size,
                              hipStream_t stream) {
    const float* x = (const float*)d_in[0];   // [4,4096,2048] f32
    const float* W = (const float*)d_in[1];   // [8,2048] f32

    float* out     = (float*)d_out;
    float* out_g   = out;                     // [16384,2] f32 gate probs
    int*   out_i   = (int*)(out + 2 * N_TOK); // [16384,2] i32 top-2 indices
    float* out_aux = out + 4 * N_TOK;         // scalar aux loss

    float* ws_imp  = (float*)d_ws;            // [8] importance sums
    float* ws_load = ws_imp + N_EXP;          // [8] top-1 counts

    hipMemsetAsync(d_ws, 0, 2 * N_EXP * sizeof(float), stream);
    top2gate_kernel<<<N_TOK / 16, 256, 0, stream>>>(x, W, out_g, out_i,
                                                    ws_imp, ws_load);
    top2gate_finalize<<<1, 32, 0, stream>>>(ws_imp, ws_load, out_aux);
}